// DenseGraphAttentionLayer_59914793780039
// MI455X (gfx1250) — compile-verified
//
#include <hip/hip_runtime.h>
#include <math.h>

// ---------------------------------------------------------------------------
// Dense GAT, fused, fp32 WMMA (V_WMMA_F32_16X16X4_F32) on gfx1250 / MI455X.
//   B=8, N=1024, I=64, O=32, H=4
// Kernel 1: projected[b,h] = x[b] @ W[h] + bias; src/dst scores.
//           proj stored PAIR-SWIZZLED: proj[bh][(j>>1)*64 + col*2 + (j&1)]
//           so kernel 2's WMMA B tiles are single ds_load_b64's.
// Kernel 2: fused masked softmax + aggregation; projected tile staged into
//           LDS via gfx1250 async global->LDS loads (ASYNCcnt).
// ---------------------------------------------------------------------------

typedef __attribute__((ext_vector_type(2))) float v2f;
typedef __attribute__((ext_vector_type(8))) float v8f;

#define B_ 8
#define N_ 1024
#define I_ 64
#define O_ 32
#define H_ 4
#define NEG_INF_ (-1e30f)

__device__ __forceinline__ float lrelu02(float v) {
    return v > 0.0f ? v : 0.2f * v;
}

// ---------------------------------------------------------------------------
// Kernel 1: projection + attention scores.
// One wave per 16-row tile of one (b,h). 2048 waves -> 256 blocks x 256 thr.
// ---------------------------------------------------------------------------
__global__ __launch_bounds__(256) void gat_project(
    const float* __restrict__ x,        // [B,N,I]
    const float* __restrict__ weight,   // [H,I,O]
    const float* __restrict__ bias,     // [H,O]
    const float* __restrict__ attn_src, // [H,O]
    const float* __restrict__ attn_dst, // [H,O]
    float* __restrict__ proj,           // [B*H, N*O] pair-swizzled
    float* __restrict__ srcS,           // [B*H, N]
    float* __restrict__ dstS)           // [B*H, N]
{
    const int wave = (blockIdx.x * blockDim.x + threadIdx.x) >> 5;
    const int lane = threadIdx.x & 31;
    const int bh   = wave >> 6;            // 64 row-tiles per (b,h)
    const int trow = (wave & 63) << 4;     // 16 rows per tile
    const int b    = bh >> 2;
    const int h    = bh & 3;
    const int col  = lane & 15;
    const int half = lane >> 4;
    const int arow = trow + col;           // A-matrix row held by this lane

    v8f acc0 = {};
    v8f acc1 = {};
    const float* xrow = x + ((size_t)b * N_ + arow) * I_;
    const float* Wh   = weight + (size_t)h * I_ * O_;

    #pragma unroll
    for (int kb = 0; kb < I_; kb += 4) {
        const int k0 = kb + 2 * half;      // lanes 16-31 supply K=2,3
        v2f a = *(const v2f*)(xrow + k0);  // 8B aligned (k0 even)
        v2f b0, b1;
        b0.x = Wh[k0 * O_ + col];
        b0.y = Wh[(k0 + 1) * O_ + col];
        b1.x = Wh[k0 * O_ + col + 16];
        b1.y = Wh[(k0 + 1) * O_ + col + 16];
        acc0 = __builtin_amdgcn_wmma_f32_16x16x4_f32(false, a, false, b0,
                                                     (short)0, acc0, false, false);
        acc1 = __builtin_amdgcn_wmma_f32_16x16x4_f32(false, a, false, b1,
                                                     (short)0, acc1, false, false);
    }

    const float bs0 = bias[h * O_ + col];
    const float bs1 = bias[h * O_ + col + 16];
    const float aS0 = attn_src[h * O_ + col];
    const float aS1 = attn_src[h * O_ + col + 16];
    const float aD0 = attn_dst[h * O_ + col];
    const float aD1 = attn_dst[h * O_ + col + 16];

    float ps[8], pd[8];
    #pragma unroll
    for (int r = 0; r < 8; ++r) {
        const float c0 = acc0[r] + bs0;
        const float c1 = acc1[r] + bs1;
        acc0[r] = c0;
        acc1[r] = c1;
        ps[r] = c0 * aS0 + c1 * aS1;   // partial src score (this lane's 2 cols)
        pd[r] = c0 * aD0 + c1 * aD1;   // partial dst score
    }
    // Reduce across the 16 lanes of the same half (offsets < 16 stay in-half).
    #pragma unroll
    for (int off = 1; off < 16; off <<= 1) {
        #pragma unroll
        for (int r = 0; r < 8; ++r) {
            ps[r] += __shfl_xor(ps[r], off, 32);
            pd[r] += __shfl_xor(pd[r], off, 32);
        }
    }

    // Pair-swizzled store: element (row, c) -> (row>>1)*64 + c*2 + (row&1).
    float* pb = proj + (size_t)bh * N_ * O_;
    #pragma unroll
    for (int r = 0; r < 8; ++r) {
        const int row = trow + r + 8 * half;  // C-layout row for this lane
        pb[(row >> 1) * 64 + col * 2 + (row & 1)]        = acc0[r];
        pb[(row >> 1) * 64 + (col + 16) * 2 + (row & 1)] = acc1[r];
    }
    if (col == 0) {
        #pragma unroll
        for (int r = 0; r < 8; ++r) {
            const int rowC = r + 8 * half;
            srcS[(size_t)bh * N_ + trow + rowC] = ps[r];
            dstS[(size_t)bh * N_ + trow + rowC] = pd[r];
        }
    }
}

// ---------------------------------------------------------------------------
// Kernel 2: fused masked softmax + aggregation.
// Block = (b,h, 128-dst-row segment); 256 threads (8 waves x 16 dst rows).
// LDS: pair-swizzled projected[b,h] (128 KB) + (src_score, gain*prior) (8 KB).
// ---------------------------------------------------------------------------
__global__ __launch_bounds__(256) void gat_attend(
    const float* __restrict__ proj,  // [B*H, N*O] pair-swizzled
    const float* __restrict__ srcS,  // [B*H, N]
    const float* __restrict__ dstS,  // [B*H, N]
    const int*   __restrict__ adj,   // [N, N]
    const float* __restrict__ prior, // [B, N]
    const float* __restrict__ beta,  // [H]
    float* __restrict__ out)         // [B, N, H*O]
{
    extern __shared__ float smem[];
    float*  Ptile = smem;                       // N_*O_ floats (128 KB)
    float2* SG    = (float2*)(smem + N_ * O_);  // N_ float2 (8 KB)

    const int bh   = blockIdx.x >> 3;
    const int seg  = blockIdx.x & 7;
    const int b    = bh >> 2;
    const int h    = bh & 3;
    const int t    = threadIdx.x;
    const int wave = t >> 5;
    const int lane = t & 31;
    const int col  = lane & 15;
    const int half = lane >> 4;

    // gain = softplus(beta[h])
    const float bt   = beta[h];
    const float gain = (bt > 20.0f) ? bt : log1pf(__expf(bt));

    // ---- Stage projected[b,h] into LDS with async global->LDS B128 --------
    {
        const float* gbase = proj + (size_t)bh * N_ * O_;
        const unsigned lds_base = (unsigned)(uintptr_t)Ptile;  // LDS byte offset
        #pragma unroll 4
        for (int it = 0; it < (N_ * O_ * 4) / (256 * 16); ++it) {
            const unsigned off = (unsigned)(it * 256 + t) * 16u;  // byte offset
            asm volatile("global_load_async_to_lds_b128 %0, %1, %2"
                         :
                         : "v"(lds_base + off), "v"(off), "s"(gbase)
                         : "memory");
        }
        for (int j = t; j < N_; j += 256) {
            float2 v;
            v.x = srcS[(size_t)bh * N_ + j];
            v.y = gain * prior[(size_t)b * N_ + j];
            SG[j] = v;
        }
        asm volatile("s_wait_asynccnt 0x0" ::: "memory");
    }
    __syncthreads();

    const int i = seg * 128 + wave * 16 + col;  // dst row owned by this lane
    const float d_i = dstS[(size_t)bh * N_ + i];
    const int* adjrow = adj + (size_t)i * N_;
    __builtin_prefetch(adjrow, 0, 3);

    // ---- Pass 1: masked row max (lane covers j = 2*half (mod 4)) ----------
    float m = NEG_INF_;
    for (int jb = 2 * half; jb < N_; jb += 4) {
        const int2   av = *(const int2*)(adjrow + jb);      // 8B aligned
        const float4 g  = *(const float4*)(&SG[jb]);        // {s0,gp0,s1,gp1}
        const float l0 = lrelu02(d_i + g.x) + g.y;
        const float l1 = lrelu02(d_i + g.z) + g.w;
        if (av.x) m = fmaxf(m, l0);
        if (av.y) m = fmaxf(m, l1);
    }
    m = fmaxf(m, __shfl_xor(m, 16, 32));
    if (m < -5e29f) m = 0.0f;   // fully-masked row -> reference uses 0

    // ---- Pass 2: exp weights in WMMA A-layout, aggregate via f32 WMMA -----
    const float2* Pp = (const float2*)Ptile;   // pair-swizzled view
    v8f acc0 = {};
    v8f acc1 = {};
    float den = 0.0f;
    for (int jb = 0; jb < N_; jb += 4) {
        const int j0 = jb + 2 * half;                       // K=0,1 / K=2,3
        const int2   av = *(const int2*)(adjrow + j0);
        const float4 g  = *(const float4*)(&SG[j0]);
        const float l0 = lrelu02(d_i + g.x) + g.y;
        const float l1 = lrelu02(d_i + g.z) + g.w;
        const float e0 = av.x ? __expf(l0 - m) : 0.0f;
        const float e1 = av.y ? __expf(l1 - m) : 0.0f;
        den += e0 + e1;

        v2f a; a.x = e0; a.y = e1;                          // A 16x4 tile
        const float2 p0 = Pp[(j0 >> 1) * 32 + col];         // {P[j0][c],P[j0+1][c]}
        const float2 p1 = Pp[(j0 >> 1) * 32 + col + 16];
        v2f b0; b0.x = p0.x; b0.y = p0.y;
        v2f b1; b1.x = p1.x; b1.y = p1.y;
        acc0 = __builtin_amdgcn_wmma_f32_16x16x4_f32(false, a, false, b0,
                                                     (short)0, acc0, false, false);
        acc1 = __builtin_amdgcn_wmma_f32_16x16x4_f32(false, a, false, b1,
                                                     (short)0, acc1, false, false);
    }
    den += __shfl_xor(den, 16, 32);          // lanes i, i+16 -> full denom_i
    den  = fmaxf(den, 1.1920929e-07f);       // eps clamp (reference)

    // ---- Epilogue: deferred softmax division + store [B,N,H*O] ------------
    const int rowbase = seg * 128 + wave * 16;
    #pragma unroll
    for (int r = 0; r < 8; ++r) {
        const int rl  = r + 8 * half;        // local row held in C VGPR r
        const float dd = __shfl(den, rl, 32);
        const float inv = 1.0f / dd;
        const int row = rowbase + rl;
        const size_t ob = ((size_t)b * N_ + row) * (H_ * O_) + (size_t)h * O_;
        out[ob + col]      = acc0[r] * inv;
        out[ob + col + 16] = acc1[r] * inv;
    }
}

// ---------------------------------------------------------------------------
extern "C" void kernel_launch(void* const* d_in, const int* in_sizes, int n_in,
                              void* d_out, int out_size, void* d_ws, size_t ws_size,
                              hipStream_t stream) {
    const float* x         = (const float*)d_in[0];
    const int*   adj       = (const int*)d_in[1];
    const float* prior     = (const float*)d_in[2];
    const float* beta      = (const float*)d_in[3];
    const float* weight    = (const float*)d_in[4];
    const float* attn_src  = (const float*)d_in[5];
    const float* attn_dst  = (const float*)d_in[6];
    const float* bias      = (const float*)d_in[7];
    float* out = (float*)d_out;

    float* ws   = (float*)d_ws;
    float* proj = ws;                                   // B*H*N*O floats (4 MB)
    float* srcS = proj + (size_t)B_ * H_ * N_ * O_;     // B*H*N floats
    float* dstS = srcS + (size_t)B_ * H_ * N_;          // B*H*N floats

    // Kernel 1: 2048 waves (one per 16-row tile) -> 256 blocks x 256 threads.
    gat_project<<<dim3(256), dim3(256), 0, stream>>>(
        x, weight, bias, attn_src, attn_dst, proj, srcS, dstS);

    // Kernel 2: B*H*8 = 256 blocks; dynamic LDS = 128KB proj + 8KB (s,gp).
    const size_t lds_bytes = (size_t)N_ * O_ * sizeof(float) + (size_t)N_ * sizeof(float2);
    gat_attend<<<dim3(B_ * H_ * 8), dim3(256), lds_bytes, stream>>>(
        proj, srcS, dstS, adj, prior, beta, out);
}